// GCN_27616639713834
// MI455X (gfx1250) — compile-verified
//
#include <hip/hip_runtime.h>
#include <hip/hip_bf16.h>
#include <math.h>

typedef float v2f __attribute__((ext_vector_type(2)));
typedef float v8f __attribute__((ext_vector_type(8)));

#define D_HID 256
#define N_CLS 40
#define KB    66            // padded dwords per k-chunk fragment block (66 ≡ 2 mod 64)
#define LDS_BYTES (4 * 64 * KB * 4)

// ---------------- CSR build ----------------

__global__ void k_count(const long long* __restrict__ ei, int* __restrict__ cnt, int E) {
    int e = blockIdx.x * blockDim.x + threadIdx.x;
    if (e < E) {
        int d = (int)ei[(size_t)E + e];
        atomicAdd(&cnt[d], 1);
    }
}

__global__ __launch_bounds__(1024) void k_scan(const int* __restrict__ cnt,
                                               int* __restrict__ rs, int n) {
    __shared__ int part[1024];
    int t = threadIdx.x;
    int CH = (n + 1023) >> 10;
    int lo = t * CH, hi = lo + CH; if (hi > n) hi = n;
    int sum = 0;
    for (int i = lo; i < hi; ++i) sum += cnt[i];
    part[t] = sum;
    __syncthreads();
    for (int off = 1; off < 1024; off <<= 1) {
        int v = (t >= off) ? part[t - off] : 0;
        __syncthreads();
        part[t] += v;
        __syncthreads();
    }
    int run = part[t] - sum;
    for (int i = lo; i < hi; ++i) { rs[i] = run; run += cnt[i]; }
    if (t == 0) rs[n] = part[1023];
}

__global__ void k_dis(const int* __restrict__ cnt, float* __restrict__ dis, int n) {
    int i = blockIdx.x * blockDim.x + threadIdx.x;
    if (i < n) dis[i] = rsqrtf((float)cnt[i] + 1.0f);
}

__global__ void k_fill(const long long* __restrict__ ei, const int* __restrict__ rs,
                       int* __restrict__ fill, int* __restrict__ csrc, int E) {
    int e = blockIdx.x * blockDim.x + threadIdx.x;
    if (e < E) {
        int s = (int)ei[e];
        int d = (int)ei[(size_t)E + e];
        int pos = atomicAdd(&fill[d], 1);
        csrc[rs[d] + pos] = s;
    }
}

// ---------------- Weight pre-pack into WMMA B-fragment order ----------------
// PB[((nt*32 + kp)*32 + lane)] : float4 = { B(k=8kp, pair), B(k=8kp+4, pair) }
// where per lane: lr = lane&15 (N col), lh = lane>>4 (K half).

__global__ void k_packW(const float* __restrict__ W, float* __restrict__ PB,
                        int N, int ntiles) {
    int idx = blockIdx.x * blockDim.x + threadIdx.x;
    int total = ntiles * 32 * 32;
    if (idx >= total) return;
    int lane = idx & 31;
    int kp   = (idx >> 5) & 31;
    int nt   = idx >> 10;
    int lr = lane & 15, lh = lane >> 4;
    int ncol = nt * 16 + lr;
    int k0 = kp * 8 + 2 * lh;
    float4 v;
    if (ncol < N) {
        v.x = W[(k0    ) * N + ncol];
        v.y = W[(k0 + 1) * N + ncol];
        v.z = W[(k0 + 4) * N + ncol];
        v.w = W[(k0 + 5) * N + ncol];
    } else { v.x = 0.f; v.y = 0.f; v.z = 0.f; v.w = 0.f; }
    ((float4*)PB)[idx] = v;
}

// ---------------- GEMM: C[Mp, NP] = A[Mp,256] * W, V_WMMA_F32_16X16X4_F32 ----------------
// 64-row M block per workgroup (4 waves). Wave w covers N-tiles w, w+4, ...
// A staged in LDS in fragment-major order; B streamed as pre-packed float4 (2 WMMAs per b128).
// No bounds checks: A row index clamped, C rows/cols padded.

template<int NTILES>
__global__ __launch_bounds__(128) void k_gemm(const float* __restrict__ A,
                                              const float* __restrict__ PB,
                                              float* __restrict__ C, int M) {
    constexpr int NP = NTILES * 16;
    extern __shared__ float sA[];                 // 4 mt * 64 kc * KB dwords
    int mbase = blockIdx.x * 64;

    // Stage A: coalesced b128 loads, conflict-free ds_store_b64 pairs.
    for (int idx = threadIdx.x; idx < 64 * 64; idx += 128) {
        int r  = idx >> 6;
        int c4 = (idx & 63) << 2;
        int rg = mbase + r; rg = rg < M ? rg : (M - 1);
        float4 va = *(const float4*)(A + (size_t)rg * 256 + c4);
        int mt = r >> 4, kc = c4 >> 2, m = r & 15;
        float* dst = sA + (mt * 64 + kc) * KB + m * 2;
        v2f lo; lo.x = va.x; lo.y = va.y;
        v2f hi; hi.x = va.z; hi.y = va.w;
        *(v2f*)(dst)      = lo;                   // lane m   (K pair 0,1)
        *(v2f*)(dst + 32) = hi;                   // lane m+16 (K pair 2,3)
    }
    __syncthreads();

    int wave = threadIdx.x >> 5;
    int lane = threadIdx.x & 31;
    int lr = lane & 15, lh = lane >> 4;
    const float* sa = sA + lane * 2;

    for (int nt = wave; nt < NTILES; nt += 4) {
        const float4* pb = (const float4*)PB + (nt << 10) + lane;
        v8f ac0 = {}, ac1 = {}, ac2 = {}, ac3 = {};
        #pragma unroll 4
        for (int kp = 0; kp < 32; ++kp) {
            float4 bq = pb[kp << 5];
            v2f b0; b0.x = bq.x; b0.y = bq.y;
            v2f b1; b1.x = bq.z; b1.y = bq.w;
            int kc = kp * 2;
            v2f a;
            #define WM_STEP(MT, ACC)                                                   \
                a = *(const v2f*)(sa + ((MT) * 64 + kc) * KB);                         \
                ACC = __builtin_amdgcn_wmma_f32_16x16x4_f32(                            \
                    false, a, false, b0, (short)0, ACC, false, false);                  \
                a = *(const v2f*)(sa + ((MT) * 64 + kc + 1) * KB);                      \
                ACC = __builtin_amdgcn_wmma_f32_16x16x4_f32(                            \
                    false, a, false, b1, (short)0, ACC, false, false);
            WM_STEP(0, ac0)
            WM_STEP(1, ac1)
            WM_STEP(2, ac2)
            WM_STEP(3, ac3)
            #undef WM_STEP
        }
        int ncol = nt * 16 + lr;
        int rowb = mbase + lh * 8;
        #pragma unroll
        for (int v = 0; v < 8; ++v) C[(size_t)(rowb      + v) * NP + ncol] = ac0[v];
        #pragma unroll
        for (int v = 0; v < 8; ++v) C[(size_t)(rowb + 16 + v) * NP + ncol] = ac1[v];
        #pragma unroll
        for (int v = 0; v < 8; ++v) C[(size_t)(rowb + 32 + v) * NP + ncol] = ac2[v];
        #pragma unroll
        for (int v = 0; v < 8; ++v) C[(size_t)(rowb + 48 + v) * NP + ncol] = ac3[v];
    }
}

// ---------------- Pull-style aggregation, D=256, fused bias+ReLU ----------------

__global__ __launch_bounds__(64) void k_agg256(const float* __restrict__ H,
                                               const int* __restrict__ rs,
                                               const int* __restrict__ csrc,
                                               const float* __restrict__ dis,
                                               const float* __restrict__ bias,
                                               float* __restrict__ out) {
    int node = blockIdx.x;
    int t = threadIdx.x;
    float di = dis[node];
    float4 hv = ((const float4*)(H + (size_t)node * 256))[t];
    float w = di * di;
    float ax = hv.x * w, ay = hv.y * w, az = hv.z * w, aw = hv.w * w;
    int e0 = rs[node], e1 = rs[node + 1];
    for (int e = e0; e < e1; ++e) {
        int s = csrc[e];
        float ww = dis[s] * di;
        float4 v = ((const float4*)(H + (size_t)s * 256))[t];
        ax += v.x * ww; ay += v.y * ww; az += v.z * ww; aw += v.w * ww;
    }
    float4 b = ((const float4*)bias)[t];
    ax = fmaxf(ax + b.x, 0.0f); ay = fmaxf(ay + b.y, 0.0f);
    az = fmaxf(az + b.z, 0.0f); aw = fmaxf(aw + b.w, 0.0f);
    float4 o; o.x = ax; o.y = ay; o.z = az; o.w = aw;
    ((float4*)(out + (size_t)node * 256))[t] = o;
}

// ---------------- Final layer: aggregation (padded D=48, valid 40) + bias + log_softmax ----------------

__global__ __launch_bounds__(32) void k_agg40_lsm(const float* __restrict__ H,
                                                  const int* __restrict__ rs,
                                                  const int* __restrict__ csrc,
                                                  const float* __restrict__ dis,
                                                  const float* __restrict__ bias,
                                                  float* __restrict__ out) {
    int node = blockIdx.x;
    int lane = threadIdx.x;
    int c0 = lane * 2;
    bool ok = c0 < N_CLS;
    float di = dis[node];
    float ax = 0.0f, ay = 0.0f;
    if (ok) {
        float2 hv = *(const float2*)(H + (size_t)node * 48 + c0);
        float w = di * di;
        ax = hv.x * w; ay = hv.y * w;
    }
    int e0 = rs[node], e1 = rs[node + 1];
    for (int e = e0; e < e1; ++e) {
        int s = csrc[e];
        float ww = dis[s] * di;
        if (ok) {
            float2 v = *(const float2*)(H + (size_t)s * 48 + c0);
            ax += v.x * ww; ay += v.y * ww;
        }
    }
    if (ok) {
        float2 b = *(const float2*)(bias + c0);
        ax += b.x; ay += b.y;
    }
    float m = ok ? fmaxf(ax, ay) : -INFINITY;
    #pragma unroll
    for (int off = 16; off >= 1; off >>= 1) m = fmaxf(m, __shfl_xor(m, off));
    float s = ok ? (__expf(ax - m) + __expf(ay - m)) : 0.0f;
    #pragma unroll
    for (int off = 16; off >= 1; off >>= 1) s += __shfl_xor(s, off);
    float lse = m + __logf(s);
    if (ok) {
        out[(size_t)node * N_CLS + c0]     = ax - lse;
        out[(size_t)node * N_CLS + c0 + 1] = ay - lse;
    }
}

// ---------------- Launch ----------------

extern "C" void kernel_launch(void* const* d_in, const int* in_sizes, int n_in,
                              void* d_out, int out_size, void* d_ws, size_t ws_size,
                              hipStream_t stream) {
    const float*     x  = (const float*)d_in[0];
    const long long* ei = (const long long*)d_in[1];   // int64 [2, E]
    const float*     W0 = (const float*)d_in[2];
    const float*     b0 = (const float*)d_in[3];
    const float*     W1 = (const float*)d_in[4];
    const float*     b1 = (const float*)d_in[5];
    const float*     W2 = (const float*)d_in[6];
    const float*     b2 = (const float*)d_in[7];
    float* outp = (float*)d_out;

    int n = in_sizes[0] / D_HID;          // 50000
    int E = in_sizes[1] / 2;              // 800000
    int Mp = (n + 63) & ~63;              // rows padded to 64 for guard-free GEMM

    char* ws = (char*)d_ws;
    size_t off = 0;
    auto alloc = [&](size_t bytes) -> void* {
        void* p = ws + off;
        off += (bytes + 255) & ~(size_t)255;
        return p;
    };
    float* HA   = (float*)alloc((size_t)Mp * 256 * 4);
    float* HB   = (float*)alloc((size_t)Mp * 256 * 4);
    float* LG   = (float*)alloc((size_t)Mp * 48 * 4);
    float* PB0  = (float*)alloc((size_t)16 * 32 * 32 * 16);
    float* PB1  = (float*)alloc((size_t)16 * 32 * 32 * 16);
    float* PB2  = (float*)alloc((size_t)3  * 32 * 32 * 16);
    int*   cnt  = (int*)alloc((size_t)n * 4);
    int*   fill = (int*)alloc((size_t)n * 4);
    int*   rs   = (int*)alloc((size_t)(n + 1) * 4);
    int*   csrc = (int*)alloc((size_t)E * 4);
    float* dis  = (float*)alloc((size_t)n * 4);

    hipMemsetAsync(cnt,  0, (size_t)n * 4, stream);
    hipMemsetAsync(fill, 0, (size_t)n * 4, stream);

    int eb = (E + 255) / 256;
    int nb = (n + 255) / 256;
    k_count<<<eb, 256, 0, stream>>>(ei, cnt, E);
    k_scan<<<1, 1024, 0, stream>>>(cnt, rs, n);
    k_dis<<<nb, 256, 0, stream>>>(cnt, dis, n);
    k_fill<<<eb, 256, 0, stream>>>(ei, rs, fill, csrc, E);

    k_packW<<<(16 * 1024 + 255) / 256, 256, 0, stream>>>(W0, PB0, D_HID, 16);
    k_packW<<<(16 * 1024 + 255) / 256, 256, 0, stream>>>(W1, PB1, D_HID, 16);
    k_packW<<<(3  * 1024 + 255) / 256, 256, 0, stream>>>(W2, PB2, N_CLS, 3);

    int gb = Mp / 64;
    // Layer 0
    k_gemm<16><<<gb, 128, LDS_BYTES, stream>>>(x,  PB0, HA, n);
    k_agg256<<<n, 64, 0, stream>>>(HA, rs, csrc, dis, b0, HB);
    // Layer 1
    k_gemm<16><<<gb, 128, LDS_BYTES, stream>>>(HB, PB1, HA, n);
    k_agg256<<<n, 64, 0, stream>>>(HA, rs, csrc, dis, b1, HB);
    // Layer 2 (padded N=48) + fused log_softmax
    k_gemm<3><<<gb, 128, LDS_BYTES, stream>>>(HB, PB2, LG, n);
    k_agg40_lsm<<<n, 32, 0, stream>>>(LG, rs, csrc, dis, b2, outp);
}